// PointCloud_network_equiv_68521908240510
// MI455X (gfx1250) — compile-verified
//
#include <hip/hip_runtime.h>

// Problem constants (from reference): B=4, N=2048, S=256, PART=24, F=C_FEAT*NA=128*60=7680
#define B_ 4
#define N_ 2048
#define S_ 256
#define PART_ 24
#define F_ 7680
#define MPAD_ 32   // pad 24 part rows -> 32 (two 16-row WMMA tiles)

typedef __attribute__((ext_vector_type(2))) float v2f;
typedef __attribute__((ext_vector_type(8))) float v8f;

// ---------------------------------------------------------------------------
// Kernel 0: zero the padded scatter matrix A (B x 32 x 256)
// ---------------------------------------------------------------------------
__global__ __launch_bounds__(256) void k_zero_A(float* __restrict__ A) {
    A[(size_t)blockIdx.x * 256 + threadIdx.x] = 0.0f;   // grid = B*MPAD_*S_/256 = 128
}

// ---------------------------------------------------------------------------
// Kernel 1: denom[b,p] = max(sum_n |part_seg[b,n,p]|, 1e-12)
// one block per (b,p); 96 blocks x 256 threads
// ---------------------------------------------------------------------------
__global__ __launch_bounds__(256) void k_part_denom(const float* __restrict__ part_seg,
                                                    float* __restrict__ denom) {
    const int b = blockIdx.x / PART_;
    const int p = blockIdx.x % PART_;
    float s = 0.0f;
    for (int n = threadIdx.x; n < N_; n += 256)
        s += fabsf(part_seg[((size_t)b * N_ + n) * PART_ + p]);
    __shared__ float red[256];
    red[threadIdx.x] = s;
    __syncthreads();
    for (int st = 128; st > 0; st >>= 1) {
        if (threadIdx.x < st) red[threadIdx.x] += red[threadIdx.x + st];
        __syncthreads();
    }
    if (threadIdx.x == 0) denom[b * PART_ + p] = fmaxf(red[0], 1e-12f);
}

// ---------------------------------------------------------------------------
// Kernel 2: 3-NN from each dense point to the 256 sparse points + inv-dist weights
// one thread per (b,n); xyz2 slice staged in LDS. grid = B*N/256 = 32 blocks
// ---------------------------------------------------------------------------
__global__ __launch_bounds__(256) void k_knn(const float* __restrict__ xyz1,
                                             const float* __restrict__ xyz2,
                                             int* __restrict__ idx_out,
                                             float* __restrict__ w_out) {
    __shared__ float sx[S_], sy[S_], sz[S_];
    const int t = blockIdx.x * 256 + threadIdx.x;     // global (b,n) id
    const int b = t / N_;                             // constant within block (256 | 2048)
    const float* x2 = xyz2 + (size_t)b * 3 * S_;      // (3, S) layout
    sx[threadIdx.x] = x2[threadIdx.x];
    sy[threadIdx.x] = x2[S_ + threadIdx.x];
    sz[threadIdx.x] = x2[2 * S_ + threadIdx.x];
    __syncthreads();

    const float px = xyz1[(size_t)t * 3 + 0];
    const float py = xyz1[(size_t)t * 3 + 1];
    const float pz = xyz1[(size_t)t * 3 + 2];

    float d0 = 3.402823466e+38f, d1 = d0, d2 = d0;
    int   i0 = 0, i1 = 0, i2 = 0;
    for (int s = 0; s < S_; ++s) {
        const float dx = px - sx[s], dy = py - sy[s], dz = pz - sz[s];
        const float d = dx * dx + dy * dy + dz * dz;
        if (d < d0)      { d2 = d1; i2 = i1; d1 = d0; i1 = i0; d0 = d; i0 = s; }
        else if (d < d1) { d2 = d1; i2 = i1; d1 = d;  i1 = s; }
        else if (d < d2) { d2 = d;  i2 = s; }
    }
    float w0 = 1.0f / (d0 + 1e-8f);
    float w1 = 1.0f / (d1 + 1e-8f);
    float w2 = 1.0f / (d2 + 1e-8f);
    const float inv = 1.0f / (w0 + w1 + w2);
    idx_out[(size_t)t * 3 + 0] = i0;
    idx_out[(size_t)t * 3 + 1] = i1;
    idx_out[(size_t)t * 3 + 2] = i2;
    w_out[(size_t)t * 3 + 0] = w0 * inv;
    w_out[(size_t)t * 3 + 1] = w1 * inv;
    w_out[(size_t)t * 3 + 2] = w2 * inv;
}

// ---------------------------------------------------------------------------
// Kernel 3: build scatter matrix A[b, p, s] = sum_{n,k} wn[b,n,p]*w[b,n,k]*[idx==s]
// Implemented as a deterministic GATHER: block = (b,p), thread = s, serial loop n.
// Loads are wave-uniform (scalarized); fixed accumulation order -> bitwise stable.
// ---------------------------------------------------------------------------
__global__ __launch_bounds__(256) void k_scatter(const float* __restrict__ part_seg,
                                                 const float* __restrict__ denom,
                                                 const int* __restrict__ idx,
                                                 const float* __restrict__ w,
                                                 float* __restrict__ A) {
    const int b = blockIdx.x / PART_;
    const int p = blockIdx.x % PART_;
    const int s = threadIdx.x;
    const float invd = 1.0f / denom[b * PART_ + p];
    float acc = 0.0f;
    const size_t nb = (size_t)b * N_;
    for (int n = 0; n < N_; ++n) {
        const float wn = part_seg[(nb + n) * PART_ + p] * invd;
        const size_t base = (nb + n) * 3;
        const int j0 = idx[base + 0], j1 = idx[base + 1], j2 = idx[base + 2];
        const float a0 = wn * w[base + 0];
        const float a1 = wn * w[base + 1];
        const float a2 = wn * w[base + 2];
        if (j0 == s) acc += a0;
        if (j1 == s) acc += a1;
        if (j2 == s) acc += a2;
    }
    A[((size_t)b * MPAD_ + p) * S_ + s] = acc;   // rows 24..31 stay zero (k_zero_A)
}

// ---------------------------------------------------------------------------
// Kernel 4: out[b,p,f] = sum_s A[b,p,s] * points2[b,f,s]
// Full-precision f32 WMMA (V_WMMA_F32_16X16X4_F32). One wave per (b, 16-col f-tile),
// computing both 16-row M tiles so the points2 stream (dominant traffic) is read once.
// grid = 240 blocks x 256 thr = 1920 waves = B * (F/16).
// ---------------------------------------------------------------------------
__global__ __launch_bounds__(256) void k_gemm_wmma(const float* __restrict__ A,
                                                   const float* __restrict__ points2,
                                                   float* __restrict__ out) {
    const int wave  = (int)((blockIdx.x * 256 + threadIdx.x) >> 5);
    const int lane  = threadIdx.x & 31;
    const int half  = lane >> 4;        // 0: K pair {0,1}; 1: K pair {2,3}
    const int lidx  = lane & 15;        // row (A) / col (B) within the 16x16 tile
    const int b     = wave / (F_ / 16);
    const int ftile = wave % (F_ / 16);

    const float* Ab   = A + (size_t)b * MPAD_ * S_;
    const float* Ar0  = Ab + (size_t)lidx * S_;          // M tile 0: rows 0..15
    const float* Ar1  = Ab + (size_t)(16 + lidx) * S_;   // M tile 1: rows 16..31
    const float* Brow = points2 + ((size_t)b * F_ + ftile * 16 + lidx) * S_;

    v8f c0 = {};
    v8f c1 = {};
    for (int k0 = 0; k0 < S_; k0 += 4) {
        const int ka = k0 + half * 2;
        const float2 bv  = *(const float2*)(Brow + ka);
        const float2 a0v = *(const float2*)(Ar0 + ka);
        const float2 a1v = *(const float2*)(Ar1 + ka);
        v2f bf; bf.x = bv.x;  bf.y = bv.y;
        v2f a0; a0.x = a0v.x; a0.y = a0v.y;
        v2f a1; a1.x = a1v.x; a1.y = a1v.y;
        c0 = __builtin_amdgcn_wmma_f32_16x16x4_f32(false, a0, false, bf, (short)0, c0, false, false);
        c1 = __builtin_amdgcn_wmma_f32_16x16x4_f32(false, a1, false, bf, (short)0, c1, false, false);
    }

    const int fout = ftile * 16 + lidx;
    const size_t obase = (size_t)b * PART_ * F_;
#pragma unroll
    for (int r = 0; r < 8; ++r) {
        const int m0 = r + half * 8;               // 0..15, all valid
        out[obase + (size_t)m0 * F_ + fout] = c0[r];
        const int m1 = 16 + r + half * 8;          // 16..31, keep 16..23
        if (m1 < PART_)
            out[obase + (size_t)m1 * F_ + fout] = c1[r];
    }
}

// ---------------------------------------------------------------------------
extern "C" void kernel_launch(void* const* d_in, const int* in_sizes, int n_in,
                              void* d_out, int out_size, void* d_ws, size_t ws_size,
                              hipStream_t stream) {
    const float* xyz1     = (const float*)d_in[0];   // (B, N, 3)
    const float* xyz2     = (const float*)d_in[1];   // (B, 3, S)
    const float* points2  = (const float*)d_in[2];   // (B, F, S)
    const float* part_seg = (const float*)d_in[3];   // (B, N, PART)
    float* out = (float*)d_out;                      // (B, PART, F) flat

    // Workspace layout (256B-aligned offsets)
    char* ws = (char*)d_ws;
    float* A     = (float*)(ws);                         // B*32*256 floats = 131072 B
    float* denom = (float*)(ws + 131072);                // 96 floats (pad 512 B)
    int*   idx   = (int*)  (ws + 131584);                // B*N*3 ints  = 98304 B
    float* w     = (float*)(ws + 229888);                // B*N*3 floats = 98304 B
    (void)in_sizes; (void)n_in; (void)out_size; (void)ws_size;

    k_zero_A     <<<(B_ * MPAD_ * S_) / 256, 256, 0, stream>>>(A);
    k_part_denom <<<B_ * PART_,              256, 0, stream>>>(part_seg, denom);
    k_knn        <<<(B_ * N_) / 256,         256, 0, stream>>>(xyz1, xyz2, idx, w);
    k_scatter    <<<B_ * PART_,              256, 0, stream>>>(part_seg, denom, idx, w, A);
    k_gemm_wmma  <<<(B_ * (F_ / 16)) / 8,    256, 0, stream>>>(A, points2, out);
}